// VectorQuantizerEMA_47132971106722
// MI455X (gfx1250) — compile-verified
//
#include <hip/hip_runtime.h>

// ---------------- problem constants ----------------
#define D_DIM   256
#define K_CODES 8192
#define T_TOK   16384
#define GAMMA_F 0.99f
#define EPS_F   1e-5f

// ---------------- WMMA types ----------------
typedef float v2f __attribute__((ext_vector_type(2)));
typedef float v8f __attribute__((ext_vector_type(8)));

// ---------------- tiling ----------------
#define ROWS_PER_WG 128          // 8 waves x 16-row bands
#define KP          64           // columns per pass (4 x 16-col WMMA tiles)
#define NPASS       (K_CODES / KP)      // 128
#define XS_STRIDE   260          // 256 + 4 pad: float4-aligned, conflict-free b64 reads
#define ES_STRIDE   260

// =====================================================================
// Fused  dis = ||e_k||^2 - 2 x.e_k   +  per-row argmin   (fp32 WMMA)
//   - A fragment (x band) loaded once per d-step, reused by 4 WMMAs
//   - C accumulators seeded with ||e_k||^2  ->  WMMA output IS the distance
// =====================================================================
__global__ __launch_bounds__(256)
void vq_dist_argmin(const float* __restrict__ x,
                    const float* __restrict__ emb,      // [D, K]
                    const float* __restrict__ colnorm,  // [K]
                    unsigned int* __restrict__ indices) // [T]
{
    extern __shared__ float smem[];
    float* xs  = smem;                                  // 128 x 260 : holds -2*x
    float* es0 = smem + ROWS_PER_WG * XS_STRIDE;        // 64 x 260  : e tile (transposed), buf 0
    float* es1 = es0 + KP * ES_STRIDE;                  // buf 1

    const int tid  = threadIdx.x;
    const int wave = tid >> 5;
    const int lane = tid & 31;
    const int ln   = lane & 15;       // column within 16-col tile / row within band
    const int g    = lane >> 4;       // half-wave group (K-slice select)
    const int rowBase = blockIdx.x * ROWS_PER_WG;

    // ---- stage x tile into LDS, scaled by -2 (float4, fully coalesced) ----
    for (int i = tid; i < ROWS_PER_WG * (D_DIM / 4); i += 256) {
        const int r  = i >> 6;
        const int c4 = i & 63;
        const float4 v = reinterpret_cast<const float4*>(
                             x + (size_t)(rowBase + r) * D_DIM)[c4];
        float4 w;
        w.x = -2.0f * v.x; w.y = -2.0f * v.y; w.z = -2.0f * v.z; w.w = -2.0f * v.w;
        *reinterpret_cast<float4*>(&xs[r * XS_STRIDE + c4 * 4]) = w;
    }

    // ---- e-tile loader mapping: thread covers (col en, d in [edb, edb+64)) ----
    const int en  = tid & 63;
    const int edb = (tid >> 6) * 64;

    { // prologue: pass-0 tile into es0 (store transposed: esT[n][d])
        const float* src = emb + (size_t)edb * K_CODES + en;
#pragma unroll 8
        for (int i = 0; i < 64; ++i)
            es0[en * ES_STRIDE + edb + i] = src[(size_t)i * K_CODES];
    }
    __syncthreads();

    float    best[8];
    unsigned bidx[8];
#pragma unroll
    for (int j = 0; j < 8; ++j) { best[j] = 3.4e38f; bidx[j] = 0u; }

    // A fragment base: row M = ln of this wave's band, K-slice offset 2*g
    const float* aBase = &xs[(wave * 16 + ln) * XS_STRIDE + 2 * g];

    for (int p = 0; p < NPASS; ++p) {
        float* cur = (p & 1) ? es1 : es0;
        float* nxt = (p & 1) ? es0 : es1;

        // ---- issue global loads for next e tile (overlaps the WMMAs below) ----
        if (p + 1 < NPASS) {
            const int kb1 = (p + 1) * KP;
            const float* src = emb + (size_t)edb * K_CODES + kb1 + en;
#pragma unroll 8
            for (int i = 0; i < 64; ++i)
                nxt[en * ES_STRIDE + edb + i] = src[(size_t)i * K_CODES];
        }
        if (p + 2 < NPASS) {  // prefetch two passes ahead -> global_prefetch_b8
            __builtin_prefetch(emb + (size_t)edb * K_CODES + (p + 2) * KP + en, 0, 0);
        }

        // ---- 4 x 16-col distance tiles, C seeded with ||e||^2 ----
        const int kb = p * KP;
        v8f c0, c1, c2, c3;
        {
            const float cn0 = colnorm[kb      + ln];
            const float cn1 = colnorm[kb + 16 + ln];
            const float cn2 = colnorm[kb + 32 + ln];
            const float cn3 = colnorm[kb + 48 + ln];
#pragma unroll
            for (int j = 0; j < 8; ++j) {
                c0[j] = cn0; c1[j] = cn1; c2[j] = cn2; c3[j] = cn3;
            }
        }

        const float* bBase = &cur[ln * ES_STRIDE + 2 * g];
#pragma unroll 8
        for (int d0 = 0; d0 < D_DIM; d0 += 4) {
            const v2f a  = *reinterpret_cast<const v2f*>(aBase + d0);   // 1 ds_load_b64
            const v2f b0 = *reinterpret_cast<const v2f*>(bBase + d0);
            const v2f b1 = *reinterpret_cast<const v2f*>(bBase + 16 * ES_STRIDE + d0);
            const v2f b2 = *reinterpret_cast<const v2f*>(bBase + 32 * ES_STRIDE + d0);
            const v2f b3 = *reinterpret_cast<const v2f*>(bBase + 48 * ES_STRIDE + d0);
            c0 = __builtin_amdgcn_wmma_f32_16x16x4_f32(false, a, false, b0, (short)0, c0, false, false);
            c1 = __builtin_amdgcn_wmma_f32_16x16x4_f32(false, a, false, b1, (short)0, c1, false, false);
            c2 = __builtin_amdgcn_wmma_f32_16x16x4_f32(false, a, false, b2, (short)0, c2, false, false);
            c3 = __builtin_amdgcn_wmma_f32_16x16x4_f32(false, a, false, b3, (short)0, c3, false, false);
        }

        // ---- running per-row argmin (slot j -> row j + 8*g) ----
        // tiles scanned in ascending k with strict '<' => first-min semantics
#pragma unroll
        for (int j = 0; j < 8; ++j) {
            const unsigned k0 = (unsigned)(kb + ln);
            if (c0[j] < best[j]) { best[j] = c0[j]; bidx[j] = k0;      }
            if (c1[j] < best[j]) { best[j] = c1[j]; bidx[j] = k0 + 16; }
            if (c2[j] < best[j]) { best[j] = c2[j]; bidx[j] = k0 + 32; }
            if (c3[j] < best[j]) { best[j] = c3[j]; bidx[j] = k0 + 48; }
        }
        __syncthreads();
    }

    // ---- cross-lane (16-wide) min-with-index reduction, write indices ----
#pragma unroll
    for (int j = 0; j < 8; ++j) {
        float    v  = best[j];
        unsigned bi = bidx[j];
#pragma unroll
        for (int mask = 8; mask >= 1; mask >>= 1) {
            const float    ov = __shfl_xor(v, mask, 16);
            const unsigned oi = __shfl_xor(bi, mask, 16);
            if (ov < v || (ov == v && oi < bi)) { v = ov; bi = oi; }
        }
        if (ln == 0)
            indices[rowBase + wave * 16 + j + 8 * g] = bi;
    }
}

// =====================================================================
// Small helper kernels
// =====================================================================
__global__ void vq_zero(float* __restrict__ p, size_t n) {
    for (size_t i = (size_t)blockIdx.x * 256 + threadIdx.x; i < n;
         i += (size_t)gridDim.x * 256)
        p[i] = 0.0f;
}

// colnorm[k] = sum_d e[d,k]^2
__global__ void vq_colnorm(const float* __restrict__ emb, float* __restrict__ colnorm) {
    const int k = blockIdx.x * 256 + threadIdx.x;
    float s = 0.0f;
#pragma unroll 8
    for (int d = 0; d < D_DIM; ++d) {
        const float e = emb[(size_t)d * K_CODES + k];
        s = fmaf(e, e, s);
    }
    colnorm[k] = s;
}

// quantized_st[t, d] = emb[d, idx[t]]   (numerically x + sg(q - x) == q)
__global__ void vq_gather(const float* __restrict__ emb,
                          const unsigned int* __restrict__ indices,
                          float* __restrict__ qout) {
    const int t = blockIdx.x;
    const int d = threadIdx.x;
    const unsigned k = indices[t];
    qout[(size_t)t * D_DIM + d] = emb[(size_t)d * K_CODES + k];
}

// counts[k] += 1 ; sums[k, d] += x[t, d]
__global__ void vq_scatter(const float* __restrict__ x,
                           const unsigned int* __restrict__ indices,
                           float* __restrict__ sums,
                           float* __restrict__ counts) {
    const int t = blockIdx.x;
    const int d = threadIdx.x;
    const unsigned k = indices[t];
    atomicAdd(&sums[(size_t)k * D_DIM + d], x[(size_t)t * D_DIM + d]);
    if (d == 0) atomicAdd(&counts[k], 1.0f);
}

// new_N = gamma*N + (1-gamma)*counts
__global__ void vq_newN(const float* __restrict__ N,
                        const float* __restrict__ counts,
                        float* __restrict__ N_out) {
    const int k = blockIdx.x * 256 + threadIdx.x;
    N_out[k] = GAMMA_F * N[k] + (1.0f - GAMMA_F) * counts[k];
}

// deterministic single-block sum of new_N
__global__ void vq_nsum(const float* __restrict__ N_out, float* __restrict__ nsum) {
    __shared__ float red[256];
    float s = 0.0f;
    for (int k = threadIdx.x; k < K_CODES; k += 256) s += N_out[k];
    red[threadIdx.x] = s;
    __syncthreads();
    for (int st = 128; st > 0; st >>= 1) {
        if (threadIdx.x < st) red[threadIdx.x] += red[threadIdx.x + st];
        __syncthreads();
    }
    if (threadIdx.x == 0) nsum[0] = red[0];
}

// new_m = gamma*m + (1-gamma)*sums^T ; new_emb = new_m / N_scaled[k]
__global__ void vq_final(const float* __restrict__ m,
                         const float* __restrict__ sums,
                         const float* __restrict__ N_out,
                         const float* __restrict__ nsum,
                         float* __restrict__ m_out,
                         float* __restrict__ emb_out) {
    const size_t i = (size_t)blockIdx.x * 256 + threadIdx.x;   // over D*K
    const int k = (int)(i & (K_CODES - 1));
    const int d = (int)(i >> 13);
    const float nm = GAMMA_F * m[i] + (1.0f - GAMMA_F) * sums[(size_t)k * D_DIM + d];
    m_out[i] = nm;
    const float Ns = nsum[0];
    const float scaled = (N_out[k] + EPS_F) / (Ns + (float)K_CODES * EPS_F) * Ns;
    emb_out[i] = nm / scaled;
}

// =====================================================================
// Launch
// =====================================================================
extern "C" void kernel_launch(void* const* d_in, const int* in_sizes, int n_in,
                              void* d_out, int out_size, void* d_ws, size_t ws_size,
                              hipStream_t stream) {
    const float* x   = (const float*)d_in[0];   // [16,32,32,256]
    const float* emb = (const float*)d_in[1];   // [256, 8192]
    const float* N   = (const float*)d_in[2];   // [8192]
    const float* m   = (const float*)d_in[3];   // [256, 8192]

    float* out     = (float*)d_out;
    float* q_out   = out;                                    // 4,194,304
    float* emb_out = out + (size_t)T_TOK * D_DIM;            // 2,097,152
    float* N_out   = emb_out + (size_t)D_DIM * K_CODES;      // 8,192
    float* m_out   = N_out + K_CODES;                        // 2,097,152

    float* ws       = (float*)d_ws;
    float* colnorm  = ws;                                    // K
    float* counts   = ws + K_CODES;                          // K
    float* nsum     = ws + 2 * K_CODES;                      // 1 (padded to 32)
    unsigned int* indices = (unsigned int*)(ws + 2 * K_CODES + 32);   // T
    float* sums     = ws + 2 * K_CODES + 32 + T_TOK;         // K*D

    // 1) zero accumulators (must happen every call)
    vq_zero<<<8, 256, 0, stream>>>(counts, (size_t)K_CODES);
    vq_zero<<<2048, 256, 0, stream>>>(sums, (size_t)K_CODES * D_DIM);

    // 2) ||e_k||^2
    vq_colnorm<<<K_CODES / 256, 256, 0, stream>>>(emb, colnorm);

    // 3) fused fp32-WMMA distance + argmin
    //    LDS: 128x260 (x) + 2 x 64x260 (double-buffered e) = 260 KB < 320 KB/WGP
    const size_t lds_bytes =
        (size_t)(ROWS_PER_WG * XS_STRIDE + 2 * KP * ES_STRIDE) * sizeof(float);
    vq_dist_argmin<<<T_TOK / ROWS_PER_WG, 256, lds_bytes, stream>>>(
        x, emb, colnorm, indices);

    // 4) quantized (straight-through) output
    vq_gather<<<T_TOK, D_DIM, 0, stream>>>(emb, indices, q_out);

    // 5) counts / sums scatter
    vq_scatter<<<T_TOK, D_DIM, 0, stream>>>(x, indices, sums, counts);

    // 6) EMA count update + deterministic total
    vq_newN<<<K_CODES / 256, 256, 0, stream>>>(N, counts, N_out);
    vq_nsum<<<1, 256, 0, stream>>>(N_out, nsum);

    // 7) EMA m update + Laplace-normalized embeddings
    vq_final<<<(D_DIM * K_CODES) / 256, 256, 0, stream>>>(
        m, sums, N_out, nsum, m_out, emb_out);
}